// MyModel_87522843560987
// MI455X (gfx1250) — compile-verified
//
#include <hip/hip_runtime.h>
#include <hip/hip_bf16.h>
#include <stdint.h>

// ---------------- problem constants (from reference) ----------------
#define N_TOTAL   4194304
#define NUM_IDS   262144
#define M_CAP     (NUM_IDS + 1)      // 262145 rows
#define L_CAP     128                // max run length emitted
#define OUT_TOTAL (M_CAP * (L_CAP + 1)) // 33,816,705 = ids region + padded region

#define BLOCK      256               // 8 wave32 per workgroup
#define PER_THREAD 16
#define CHUNK      (BLOCK * PER_THREAD)  // 4096 elements per block
#define NBLOCKS    (N_TOTAL / CHUNK)     // 1024
#define HALO       (L_CAP + 1)           // 129 ids of history bound any pos search
#define HPAD       132                   // halo slots padded so chunk is 16B-aligned

// low 32 bits of a generic pointer to LDS == DS address (aperture layout)
__device__ __forceinline__ uint32_t lds_addr32(const void* p) {
    return (uint32_t)(uintptr_t)p;
}

// =====================================================================
// Kernel 1: initialize output. run_ids region = -1 (int), padded = 0.0f.
// int bit pattern 0 == 0.0f, so a single int fill covers both regions.
// b128 stores via int4: this fill IS the roofline cost (~135 MB).
// =====================================================================
__global__ void rlg_fill_kernel(int* __restrict__ out, int total, int idCount) {
    int4* out4 = (int4*)out;
    const int nvec   = total >> 2;
    const int stride = gridDim.x * blockDim.x;
    int v = blockIdx.x * blockDim.x + threadIdx.x;
    for (; v < nvec; v += stride) {
        const int b = v << 2;
        int4 val;
        val.x = (b + 0 < idCount) ? -1 : 0;
        val.y = (b + 1 < idCount) ? -1 : 0;
        val.z = (b + 2 < idCount) ? -1 : 0;
        val.w = (b + 3 < idCount) ? -1 : 0;
        out4[v] = val;
    }
    const int gtid = blockIdx.x * blockDim.x + threadIdx.x;
    if (gtid < (total & 3)) {                 // scalar tail (OUT_TOTAL % 4 == 1)
        const int idx = (nvec << 2) + gtid;
        out[idx] = (idx < idCount) ? -1 : 0;
    }
}

// =====================================================================
// Kernel 2: per-block run-start counts (prefix-sum pass 1).
// =====================================================================
__global__ void rlg_count_kernel(const int* __restrict__ ids,
                                 int* __restrict__ blockCounts) {
    const int tid  = threadIdx.x;
    const int base = blockIdx.x * CHUNK + tid * PER_THREAD;

    int prev = (base == 0) ? -1 : ids[base - 1];   // -1 sentinel: elem 0 is a start
    const int4* p = (const int4*)(ids + base);     // 64B-aligned
    int c = 0;
#pragma unroll
    for (int q = 0; q < 4; ++q) {
        const int4 v = p[q];
        c += (v.x != prev) + (v.y != v.x) + (v.z != v.y) + (v.w != v.z);
        prev = v.w;
    }

    __shared__ int red[BLOCK];
    red[tid] = c;
    __syncthreads();
#pragma unroll
    for (int off = BLOCK / 2; off > 0; off >>= 1) {
        if (tid < off) red[tid] += red[tid + off];
        __syncthreads();
    }
    if (tid == 0) blockCounts[blockIdx.x] = red[0];
}

// =====================================================================
// Kernel 3: exclusive scan of 1024 block counts + total run count.
// =====================================================================
__global__ void rlg_scan_kernel(const int* __restrict__ counts,
                                int* __restrict__ offsets,
                                int* __restrict__ totalp, int n) {
    __shared__ int s[1024];
    const int tid = threadIdx.x;
    const int c = (tid < n) ? counts[tid] : 0;
    s[tid] = c;
    __syncthreads();
    for (int off = 1; off < 1024; off <<= 1) {
        const int t = (tid >= off) ? s[tid - off] : 0;
        __syncthreads();
        s[tid] += t;
        __syncthreads();
    }
    if (tid < n)      offsets[tid] = s[tid] - c;   // exclusive
    if (tid == n - 1) *totalp      = s[tid];       // num_runs
}

// =====================================================================
// Kernel 4: scatter. ids chunk + halo staged LDS-direct with CDNA5
// GLOBAL_LOAD_ASYNC_TO_LDS (ASYNCcnt), then run numbering via block scan
// and pos-in-run via incremental counter seeded by an 8-step LDS search.
// =====================================================================
__device__ __forceinline__ int rlg_run_pos(const int* __restrict__ s, int li) {
    const int v = s[li];
    int lo = li - L_CAP;              // >= 4 always (li >= HPAD = 132)
    int hi = li;                      // invariant: s[hi] == v
    while (lo < hi) {                 // first occurrence of v in window
        const int mid = (lo + hi) >> 1;
        if (s[mid] == v) hi = mid; else lo = mid + 1;
    }
    return li - hi;                   // in [0, 128]; 128 means "at least 128" -> drop
}

__global__ void rlg_scatter_kernel(const int* __restrict__ ids,
                                   const float* __restrict__ feats,
                                   const int* __restrict__ blockOffsets,
                                   const int* __restrict__ totalp,
                                   int* __restrict__ run_ids,
                                   float* __restrict__ padded) {
    __shared__ __align__(16) int s_ids[HPAD + CHUNK];   // 16.9 KB of 320 KB/WGP
    __shared__ int s_scan[BLOCK];

    const int tid        = threadIdx.x;
    const int blockStart = blockIdx.x * CHUNK;
    const int base       = blockStart + tid * PER_THREAD;

    // hint the feature stream toward cache while staging ids (global_prefetch_b8)
    __builtin_prefetch(feats + base, 0, 3);

    // --- async stage main chunk: 4096 ints as 1024 x b128, 16B-aligned both sides ---
#pragma unroll
    for (int q = 0; q < 4; ++q) {
        const int t = (q * BLOCK + tid) * 4;                   // int index in chunk
        const uint32_t ldsa = lds_addr32(&s_ids[HPAD + t]);    // 528 + 16k bytes
        const uint64_t ga   = (uint64_t)(uintptr_t)(ids + blockStart + t);
        asm volatile("global_load_async_to_lds_b128 %0, %1, off"
                     :: "v"(ldsa), "v"(ga) : "memory");
    }
    // --- async stage halo: 129 ints as b32 (address clamped; block 0 fixed up) ---
    if (tid < HALO) {
        int g = blockStart - HALO + tid;
        if (g < 0) g = 0;
        const uint32_t ldsa = lds_addr32(&s_ids[HPAD - HALO + tid]);
        const uint64_t ga   = (uint64_t)(uintptr_t)(ids + g);
        asm volatile("global_load_async_to_lds_b32 %0, %1, off"
                     :: "v"(ldsa), "v"(ga) : "memory");
    }
    asm volatile("s_wait_asynccnt 0x0" ::: "memory");  // this wave's LDS writes landed
    __syncthreads();                                   // publish all waves' staging
    if (blockStart == 0 && tid < HALO)                 // sentinel: ids[-1..] := -1
        s_ids[HPAD - HALO + tid] = -1;
    __syncthreads();

    const int l0 = HPAD + tid * PER_THREAD;   // LDS coord of this thread's first elem

    // ---- phase 1: per-thread run-start count ----
    int prev = s_ids[l0 - 1];
    int c = 0;
#pragma unroll
    for (int k = 0; k < PER_THREAD; ++k) {
        const int v = s_ids[l0 + k];
        c += (v != prev);
        prev = v;
    }

    // ---- phase 2: block-wide inclusive scan (Hillis-Steele) ----
    s_scan[tid] = c;
    __syncthreads();
    for (int off = 1; off < BLOCK; off <<= 1) {
        const int t = (tid >= off) ? s_scan[tid - off] : 0;
        __syncthreads();
        s_scan[tid] += t;
        __syncthreads();
    }
    const int ex = s_scan[tid] - c;                       // in-block starts before this thread

    const int lastSeg = *totalp - 1;                      // final run is never emitted
    const int segBase = blockOffsets[blockIdx.x] + ex - 1;

    // ---- phase 3: emit ----
    prev = s_ids[l0 - 1];
    int pos;
    {
        const int v0 = s_ids[l0];
        pos = (v0 != prev) ? 0 : rlg_run_pos(s_ids, l0);  // seed (one search max)
    }
    int inc = 0;
#pragma unroll
    for (int k = 0; k < PER_THREAD; ++k) {
        const int v      = s_ids[l0 + k];
        const bool start = (v != prev);
        if (k) pos = start ? 0 : pos + 1;
        inc += start ? 1 : 0;
        const int seg = segBase + inc;
        if (seg < lastSeg && pos < L_CAP) {
            padded[seg * L_CAP + pos] = feats[base + k];
            if (start) run_ids[seg] = v;   // start => pos==0, condition already satisfied
        }
        prev = v;
    }
}

// =====================================================================
// host-side launcher
// =====================================================================
extern "C" void kernel_launch(void* const* d_in, const int* in_sizes, int n_in,
                              void* d_out, int out_size, void* d_ws, size_t ws_size,
                              hipStream_t stream) {
    const int*   ids   = (const int*)d_in[0];
    const float* feats = (const float*)d_in[1];

    int*   run_ids = (int*)d_out;                 // first M_CAP ints (bit-exact int32)
    float* padded  = (float*)((int*)d_out + M_CAP);

    int* wsi     = (int*)d_ws;
    int* counts  = wsi;                 // [NBLOCKS]
    int* offsets = wsi + NBLOCKS;       // [NBLOCKS]
    int* totalp  = wsi + 2 * NBLOCKS;   // [1]

    // 1) dense output init (-1 ids region, 0.0f padded region) — the roofline cost
    rlg_fill_kernel<<<4096, BLOCK, 0, stream>>>((int*)d_out, OUT_TOTAL, M_CAP);

    // 2) per-chunk run-start counts
    rlg_count_kernel<<<NBLOCKS, BLOCK, 0, stream>>>(ids, counts);

    // 3) exclusive scan of chunk counts + total run count
    rlg_scan_kernel<<<1, 1024, 0, stream>>>(counts, offsets, totalp, NBLOCKS);

    // 4) grouped scatter with async-to-LDS staging
    rlg_scatter_kernel<<<NBLOCKS, BLOCK, 0, stream>>>(ids, feats, offsets, totalp,
                                                      run_ids, padded);
}